// Model_random_proto_77584289235342
// MI455X (gfx1250) — compile-verified
//
#include <hip/hip_runtime.h>
#include <math.h>

// ---- problem constants (from reference) ----
#define NN    50000
#define EE    600000
#define DH    128
#define CC    7
#define PROT  100
#define PPAD  112      // 100 padded up to 7 tiles of 16
#define EPS6  1e-6f

#define TPB_TILES 5    // row tiles per block in WMMA GEMMs (3125 = 625 * 5)

typedef float v2f __attribute__((ext_vector_type(2)));
typedef float v8f __attribute__((ext_vector_type(8)));

// ---- gfx1250 async global->LDS copy (guarded; falls back to plain LDS stores) ----
#if __has_builtin(__builtin_amdgcn_global_load_async_to_lds_b128) && \
    __has_builtin(__builtin_amdgcn_s_wait_asynccnt)
#define USE_ASYNC_LDS 1
typedef int v4i32 __attribute__((vector_size(4 * sizeof(int))));
typedef __attribute__((address_space(1))) v4i32* gv4_ptr_t;   // global int4*
typedef __attribute__((address_space(3))) v4i32* lv4_ptr_t;   // LDS int4*
#define ASYNC_WAIT(K) __builtin_amdgcn_s_wait_asynccnt(K)
#else
#define USE_ASYNC_LDS 0
#define ASYNC_WAIT(K)
#endif

// copy 8 consecutive floats (2 x b128) for this thread into the LDS tile buffer
__device__ __forceinline__ void stage_tile_row(const float* __restrict__ src,
                                               float* __restrict__ dst) {
#if USE_ASYNC_LDS
  __builtin_amdgcn_global_load_async_to_lds_b128((gv4_ptr_t)(void*)src,
                                                 (lv4_ptr_t)(void*)dst, 0, 0);
  __builtin_amdgcn_global_load_async_to_lds_b128((gv4_ptr_t)(void*)(src + 4),
                                                 (lv4_ptr_t)(void*)(dst + 4), 0, 0);
#else
  float4 v0 = ((const float4*)src)[0];
  float4 v1 = ((const float4*)src)[1];
  ((float4*)dst)[0] = v0;
  ((float4*)dst)[1] = v1;
#endif
}

// ---------------------------------------------------------------------------
// degree / normalization
// ---------------------------------------------------------------------------
__global__ void k_deg_init(float* __restrict__ deg) {
  int i = blockIdx.x * blockDim.x + threadIdx.x;
  if (i < NN) deg[i] = 1.0f;               // self loop
}

__global__ void k_deg_accum(const int* __restrict__ dst, float* __restrict__ deg) {
  int e = blockIdx.x * blockDim.x + threadIdx.x;
  if (e < EE) unsafeAtomicAdd(&deg[dst[e]], 1.0f);
}

__global__ void k_deg_to_dinv(const float* __restrict__ deg, float* __restrict__ dinv) {
  int i = blockIdx.x * blockDim.x + threadIdx.x;
  if (i < NN) {
    float d = deg[i];
    dinv[i] = (d > 0.0f) ? rsqrtf(d) : 0.0f;
  }
}

// ---------------------------------------------------------------------------
// dense GEMM  O[N,128] = X[N,128] @ W[128,128]   (f32 WMMA 16x16x4)
//  - 256 threads = 8 waves; wave w owns column tile w
//  - B panel (16 cols x 128 K) persists in 64 VGPRs per wave
//  - A row-tile (16x128) staged into LDS once per tile via async copy,
//    double buffered, shared by all 8 waves
//  - dual accumulators break the WMMA RAW chain (2x ILP)
// ---------------------------------------------------------------------------
__global__ __launch_bounds__(256) void k_gemm_x128(const float* __restrict__ X,
                                                   const float* __restrict__ W,
                                                   float* __restrict__ O) {
  __shared__ float As[2][16 * DH];          // 2 x 8KB double buffer
  const int tid  = threadIdx.x;
  const int wave = tid >> 5;                // 0..7 -> col tile
  const int lane = tid & 31;
  const int m    = lane & 15;               // A row / B,C column within tile
  const int kg   = (lane >> 4) * 2;         // K sub-offset for this half-wave
  const int col0 = wave * 16;

  // persistent B panel: one v2f per K-step of 4
  v2f B[32];
#pragma unroll
  for (int ks = 0; ks < 32; ++ks) {
    const float* wp = W + (size_t)(ks * 4 + kg) * DH + col0 + m;
    B[ks].x = wp[0];
    B[ks].y = wp[DH];
  }

  const int tile0 = blockIdx.x * TPB_TILES;

  // prologue: stage tile 0 into buffer 0
  stage_tile_row(X + (size_t)tile0 * 16 * DH + tid * 8, &As[0][tid * 8]);

  for (int t = 0; t < TPB_TILES; ++t) {
    const int buf = t & 1;
    if (t + 1 < TPB_TILES) {
      stage_tile_row(X + (size_t)(tile0 + t + 1) * 16 * DH + tid * 8,
                     &As[buf ^ 1][tid * 8]);
      ASYNC_WAIT(2);                        // previous batch (current buf) landed
    } else {
      ASYNC_WAIT(0);
    }
    __syncthreads();                        // all waves' staging of As[buf] visible

    const float* arow = &As[buf][m * DH];
    v8f c0 = {}, c1 = {};
#pragma unroll
    for (int ks = 0; ks < 32; ks += 2) {
      v2f a0, a1;
      a0.x = arow[ks * 4 + kg];
      a0.y = arow[ks * 4 + kg + 1];
      a1.x = arow[ks * 4 + 4 + kg];
      a1.y = arow[ks * 4 + 4 + kg + 1];
      c0 = __builtin_amdgcn_wmma_f32_16x16x4_f32(false, a0, false, B[ks],
                                                 (short)0, c0, false, false);
      c1 = __builtin_amdgcn_wmma_f32_16x16x4_f32(false, a1, false, B[ks + 1],
                                                 (short)0, c1, false, false);
    }

    const int row0  = (tile0 + t) * 16;
    const int mbase = (lane >> 4) * 8;
#pragma unroll
    for (int v = 0; v < 8; ++v)
      O[(size_t)(row0 + mbase + v) * DH + col0 + m] = c0[v] + c1[v];

    __syncthreads();                        // protect As[buf] before re-stage
  }
}

// ---------------------------------------------------------------------------
// sparse aggregation: acc = selfloop + scatter(hw[s]*dinv[s]*dinv[d])
// ---------------------------------------------------------------------------
__global__ void k_acc_selfloop(const float* __restrict__ hw, const float* __restrict__ dinv,
                               float* __restrict__ acc) {
  int i = blockIdx.x * blockDim.x + threadIdx.x;
  if (i < NN * DH) {
    float dv = dinv[i >> 7];                // row = i / 128
    acc[i] = hw[i] * dv * dv;
  }
}

// wave per edge: lane handles 4 consecutive features (b128 load + 4 f32 atomics)
__global__ void k_edge_scatter(const float* __restrict__ hw, const int* __restrict__ src,
                               const int* __restrict__ dst, const float* __restrict__ dinv,
                               float* __restrict__ acc) {
  const int e    = blockIdx.x * 8 + (threadIdx.x >> 5);
  const int lane = threadIdx.x & 31;
  if (e >= EE) return;
  const int s = src[e];
  const int d = dst[e];
  const float w = dinv[s] * dinv[d];
  const float4 v = ((const float4*)(hw + (size_t)s * DH))[lane];
  float* ap = acc + (size_t)d * DH + lane * 4;
  unsafeAtomicAdd(ap + 0, v.x * w);
  unsafeAtomicAdd(ap + 1, v.y * w);
  unsafeAtomicAdd(ap + 2, v.z * w);
  unsafeAtomicAdd(ap + 3, v.w * w);
}

__global__ void k_bias_relu(float* __restrict__ acc, const float* __restrict__ b) {
  int i = blockIdx.x * blockDim.x + threadIdx.x;
  if (i < NN * DH) {
    float v = acc[i] + b[i & (DH - 1)];
    acc[i] = v > 0.0f ? v : 0.0f;
  }
}

// ---------------------------------------------------------------------------
// last layer: add bias, row-L2-normalize in place, record nx = ||row|| (post-norm)
// one wave (32 lanes) per row, 8 rows per 256-thread block
// ---------------------------------------------------------------------------
__global__ void k_bias_l2norm(float* __restrict__ H, const float* __restrict__ b,
                              float* __restrict__ nx) {
  const int row  = blockIdx.x * 8 + (threadIdx.x >> 5);
  const int lane = threadIdx.x & 31;
  if (row >= NN) return;
  float* hr = H + (size_t)row * DH;

  float vals[4];
  float ss = 0.0f;
#pragma unroll
  for (int i = 0; i < 4; ++i) {
    int col = lane + 32 * i;
    vals[i] = hr[col] + b[col];
    ss += vals[i] * vals[i];
  }
#pragma unroll
  for (int off = 16; off > 0; off >>= 1) ss += __shfl_xor(ss, off, 32);
  const float qn = sqrtf(ss);

  float ss2 = 0.0f;
#pragma unroll
  for (int i = 0; i < 4; ++i) {
    float v = vals[i] / qn;
    hr[lane + 32 * i] = v;
    ss2 += v * v;
  }
#pragma unroll
  for (int off = 16; off > 0; off >>= 1) ss2 += __shfl_xor(ss2, off, 32);
  if (lane == 0) nx[row] = sqrtf(ss2);
}

// ---------------------------------------------------------------------------
// gather anchors (transposed + zero-padded to 112 cols) and anchor norms
// ---------------------------------------------------------------------------
__global__ void k_gather_anchors(const float* __restrict__ H, const int* __restrict__ prot,
                                 const float* __restrict__ nx, float* __restrict__ AT,
                                 float* __restrict__ na) {
  int idx = blockIdx.x * blockDim.x + threadIdx.x;   // DH * PPAD
  if (idx >= DH * PPAD) return;
  int k = idx / PPAD;
  int p = idx % PPAD;
  float v = 0.0f;
  if (p < PROT) v = H[(size_t)prot[p] * DH + k];
  AT[idx] = v;
  if (k == 0) na[p] = (p < PROT) ? nx[prot[p]] : 1.0f;
}

// ---------------------------------------------------------------------------
// x_rel[N,100] = (H @ anchors^T / max(nx*na,eps) + 1) * 0.5   (f32 WMMA)
// 224 threads = 7 waves (7 col tiles cover 112 padded anchors);
// persistent B panel in VGPRs, 5 row tiles per block, dual accumulators
// ---------------------------------------------------------------------------
__global__ __launch_bounds__(224) void k_cos_gemm(const float* __restrict__ H,
                                                  const float* __restrict__ AT,
                                                  const float* __restrict__ nx,
                                                  const float* __restrict__ na,
                                                  float* __restrict__ xrel) {
  const int wave = threadIdx.x >> 5;        // 0..6
  const int lane = threadIdx.x & 31;
  const int m    = lane & 15;
  const int kg   = (lane >> 4) * 2;
  const int col0 = wave * 16;

  v2f B[32];
#pragma unroll
  for (int ks = 0; ks < 32; ++ks) {
    const float* bp = AT + (size_t)(ks * 4 + kg) * PPAD + col0 + m;
    B[ks].x = bp[0];
    B[ks].y = bp[PPAD];
  }

  const int tile0 = blockIdx.x * TPB_TILES;
  const int col   = col0 + m;
  const float nac = (col < PROT) ? na[col] : 1.0f;

  for (int t = 0; t < TPB_TILES; ++t) {
    const int row0 = (tile0 + t) * 16;
    const float* hrow = H + (size_t)(row0 + m) * DH;
    v8f c0 = {}, c1 = {};
#pragma unroll
    for (int ks = 0; ks < 32; ks += 2) {
      v2f a0, a1;
      a0.x = hrow[ks * 4 + kg];
      a0.y = hrow[ks * 4 + kg + 1];
      a1.x = hrow[ks * 4 + 4 + kg];
      a1.y = hrow[ks * 4 + 4 + kg + 1];
      c0 = __builtin_amdgcn_wmma_f32_16x16x4_f32(false, a0, false, B[ks],
                                                 (short)0, c0, false, false);
      c1 = __builtin_amdgcn_wmma_f32_16x16x4_f32(false, a1, false, B[ks + 1],
                                                 (short)0, c1, false, false);
    }
    if (col < PROT) {
      const int mbase = (lane >> 4) * 8;
#pragma unroll
      for (int v = 0; v < 8; ++v) {
        int row = row0 + mbase + v;
        float denom = fmaxf(nx[row] * nac, EPS6);
        xrel[(size_t)row * PROT + col] = ((c0[v] + c1[v]) / denom + 1.0f) * 0.5f;
      }
    }
  }
}

// ---------------------------------------------------------------------------
// prototype head (tiny; plain VALU)
// ---------------------------------------------------------------------------
__global__ void k_proto_hidden(const float* __restrict__ AT, const float* __restrict__ lw1,
                               const float* __restrict__ lb1, float* __restrict__ hid) {
  int idx = blockIdx.x * blockDim.x + threadIdx.x;   // PROT * DH
  if (idx >= PROT * DH) return;
  int p = idx / DH;
  int j = idx % DH;
  float s = lb1[j];
  for (int k = 0; k < DH; ++k) s += AT[k * PPAD + p] * lw1[k * DH + j];
  hid[idx] = fmaxf(s, 0.0f);
}

__global__ void k_proto_out(const float* __restrict__ hid, const float* __restrict__ lw2,
                            const float* __restrict__ lb2, float* __restrict__ oproto) {
  int p = blockIdx.x * blockDim.x + threadIdx.x;
  if (p >= PROT) return;
  float z[CC];
#pragma unroll
  for (int c = 0; c < CC; ++c) {
    float s = lb2[c];
    for (int k = 0; k < DH; ++k) s += hid[p * DH + k] * lw2[k * CC + c];
    z[c] = s;
  }
  float mx = z[0];
#pragma unroll
  for (int c = 1; c < CC; ++c) mx = fmaxf(mx, z[c]);
  float se = 0.0f;
#pragma unroll
  for (int c = 0; c < CC; ++c) se += expf(z[c] - mx);
  float lse = mx + logf(se);
#pragma unroll
  for (int c = 0; c < CC; ++c) oproto[p * CC + c] = z[c] - lse;
}

__global__ void k_final(const float* __restrict__ xrel, const float* __restrict__ oproto,
                        float* __restrict__ out) {
  __shared__ float op[PROT * CC];
  for (int i = threadIdx.x; i < PROT * CC; i += blockDim.x) op[i] = oproto[i];
  __syncthreads();

  int r = blockIdx.x * blockDim.x + threadIdx.x;
  if (r >= NN) return;
  float z[CC] = {0.f, 0.f, 0.f, 0.f, 0.f, 0.f, 0.f};
  const float* xr = xrel + (size_t)r * PROT;
  for (int p = 0; p < PROT; ++p) {
    float xv = xr[p];
#pragma unroll
    for (int c = 0; c < CC; ++c) z[c] += xv * op[p * CC + c];
  }
  float mx = z[0];
#pragma unroll
  for (int c = 1; c < CC; ++c) mx = fmaxf(mx, z[c]);
  float se = 0.0f;
#pragma unroll
  for (int c = 0; c < CC; ++c) se += expf(z[c] - mx);
  float lse = mx + logf(se);
#pragma unroll
  for (int c = 0; c < CC; ++c) out[(size_t)r * CC + c] = z[c] - lse;
}

// ---------------------------------------------------------------------------
// host orchestration
// ---------------------------------------------------------------------------
extern "C" void kernel_launch(void* const* d_in, const int* in_sizes, int n_in,
                              void* d_out, int out_size, void* d_ws, size_t ws_size,
                              hipStream_t stream) {
  const float* x    = (const float*)d_in[0];
  const int*   edge = (const int*)d_in[1];
  const int*   src  = edge;           // edge_index[0]
  const int*   dst  = edge + EE;      // edge_index[1]
  const int*   prot = (const int*)d_in[2];
  const float* W0 = (const float*)d_in[3];
  const float* b0 = (const float*)d_in[4];
  const float* W1 = (const float*)d_in[5];
  const float* b1 = (const float*)d_in[6];
  const float* W2 = (const float*)d_in[7];
  const float* b2 = (const float*)d_in[8];
  const float* lw1 = (const float*)d_in[9];
  const float* lb1 = (const float*)d_in[10];
  const float* lw2 = (const float*)d_in[11];
  const float* lb2 = (const float*)d_in[12];

  // output layout: (out [N*C], x_rel [N*P], out_proto [P*C])
  float* out    = (float*)d_out;
  float* xrel   = out + (size_t)NN * CC;
  float* oproto = xrel + (size_t)NN * PROT;

  // workspace layout (floats)
  float* ws   = (float*)d_ws;
  float* buf0 = ws;                         ws += (size_t)NN * DH;   // gemm output
  float* buf1 = ws;                         ws += (size_t)NN * DH;   // h / acc (in place)
  float* deg  = ws;                         ws += NN;
  float* dinv = ws;                         ws += NN;
  float* nx   = ws;                         ws += NN;
  float* AT   = ws;                         ws += DH * PPAD;         // anchors^T padded
  float* na   = ws;                         ws += PPAD;
  float* hid  = ws;                         ws += PROT * DH;

  const int nb_n    = (NN + 255) / 256;
  const int nb_e    = (EE + 255) / 256;
  const int nb_nd   = (NN * DH + 255) / 256;
  const int gblocks = (NN / 16) / TPB_TILES;     // 625
  const int sblocks = (EE + 7) / 8;              // 8 edges per 256-thread block

  // degree / norm
  k_deg_init   <<<nb_n, 256, 0, stream>>>(deg);
  k_deg_accum  <<<nb_e, 256, 0, stream>>>(dst, deg);
  k_deg_to_dinv<<<nb_n, 256, 0, stream>>>(deg, dinv);

  // layer 0: x -> buf1
  k_gemm_x128   <<<gblocks, 256, 0, stream>>>(x, W0, buf0);
  k_acc_selfloop<<<nb_nd, 256, 0, stream>>>(buf0, dinv, buf1);
  k_edge_scatter<<<sblocks, 256, 0, stream>>>(buf0, src, dst, dinv, buf1);
  k_bias_relu   <<<nb_nd, 256, 0, stream>>>(buf1, b0);

  // layer 1: buf1 -> buf1
  k_gemm_x128   <<<gblocks, 256, 0, stream>>>(buf1, W1, buf0);
  k_acc_selfloop<<<nb_nd, 256, 0, stream>>>(buf0, dinv, buf1);
  k_edge_scatter<<<sblocks, 256, 0, stream>>>(buf0, src, dst, dinv, buf1);
  k_bias_relu   <<<nb_nd, 256, 0, stream>>>(buf1, b1);

  // layer 2 (no relu) + L2 normalize
  k_gemm_x128   <<<gblocks, 256, 0, stream>>>(buf1, W2, buf0);
  k_acc_selfloop<<<nb_nd, 256, 0, stream>>>(buf0, dinv, buf1);
  k_edge_scatter<<<sblocks, 256, 0, stream>>>(buf0, src, dst, dinv, buf1);
  k_bias_l2norm <<<(NN + 7) / 8, 256, 0, stream>>>(buf1, b2, nx);

  // relative representation
  k_gather_anchors<<<(DH * PPAD + 255) / 256, 256, 0, stream>>>(buf1, prot, nx, AT, na);
  k_cos_gemm      <<<gblocks, 224, 0, stream>>>(buf1, AT, nx, na, xrel);

  // prototype head + final classification
  k_proto_hidden<<<(PROT * DH + 255) / 256, 256, 0, stream>>>(AT, lw1, lb1, hid);
  k_proto_out   <<<1, 128, 0, stream>>>(hid, lw2, lb2, oproto);
  k_final       <<<nb_n, 256, 0, stream>>>(xrel, oproto, out);

  (void)in_sizes; (void)n_in; (void)out_size; (void)ws_size;
}